// ManualGATLayer_50113678410321
// MI455X (gfx1250) — compile-verified
//
#include <hip/hip_runtime.h>

#define N_NODES   50000
#define N_EDGES   800000
#define IN_DIM    128
#define OUT_DIM   128
#define N_HEADS   4
#define HEAD_DIM  32
#define EE_DIM    16
#define LN_EPS    1e-5f

typedef __attribute__((ext_vector_type(2))) float v2f;
typedef __attribute__((ext_vector_type(8))) float v8f;

// ---------------------------------------------------------------------------
// float atomic helpers
// ---------------------------------------------------------------------------
__device__ __forceinline__ void atomicAddF(float* p, float v) {
    // agent-scope relaxed fetch_add lowers to global_atomic_add_f32 on gfx1250
    __hip_atomic_fetch_add(p, v, __ATOMIC_RELAXED, __HIP_MEMORY_SCOPE_AGENT);
}

__device__ __forceinline__ void atomicMaxF(float* addr, float val) {
    // classic monotone trick: both paths only move the stored float upward
    if (val >= 0.0f) atomicMax((int*)addr, __float_as_int(val));
    else             atomicMin((unsigned int*)addr, __float_as_uint(val));
}

// ---------------------------------------------------------------------------
// Kernel 1: h = x @ W^T   via V_WMMA_F32_16X16X4_F32
//   block = 256 threads = 8 waves; each wave: 16 rows x 128 cols
//   W (64KB) staged in LDS once per block
// ---------------------------------------------------------------------------
__global__ __launch_bounds__(256) void gemm_h_kernel(
        const float* __restrict__ x, const float* __restrict__ W,
        float* __restrict__ h) {
    __shared__ __align__(16) float Wsh[OUT_DIM * IN_DIM];   // 64 KB
    {
        const float4* Wv  = (const float4*)W;
        float4*       Wsv = (float4*)Wsh;
        for (int i = threadIdx.x; i < OUT_DIM * IN_DIM / 4; i += 256)
            Wsv[i] = Wv[i];
    }
    __syncthreads();

    const int wave = threadIdx.x >> 5;
    const int lane = threadIdx.x & 31;
    const int row0 = blockIdx.x * 128 + wave * 16;

    // A-matrix lane mapping (16x4 f32): lanes 0-15 -> K={0,1}, lanes 16-31 -> K={2,3}
    const int mrow   = row0 + (lane & 15);
    const int mclamp = mrow < N_NODES ? mrow : (N_NODES - 1);  // keep EXEC all-ones
    const int khalf  = (lane >> 4) * 2;

    v8f acc[8] = {};   // 8 N-tiles of 16 cols each

    for (int k = 0; k < IN_DIM; k += 4) {
        v2f a = *(const v2f*)(x + (size_t)mclamp * IN_DIM + k + khalf);
#pragma unroll
        for (int nt = 0; nt < 8; ++nt) {
            const int ncol = nt * 16 + (lane & 15);           // B[k][n] = W[n][k]
            v2f b = *(const v2f*)(Wsh + ncol * IN_DIM + k + khalf);
            acc[nt] = __builtin_amdgcn_wmma_f32_16x16x4_f32(
                false, a, false, b, (short)0, acc[nt], false, false);
        }
    }

    // C/D layout: VGPR r -> (M=r, N=lane) lanes 0-15 ; (M=r+8, N=lane-16) lanes 16-31
    const int colbase = lane & 15;
    const int rowadd  = (lane >> 4) * 8;
#pragma unroll
    for (int nt = 0; nt < 8; ++nt) {
#pragma unroll
        for (int r = 0; r < 8; ++r) {
            const int row = row0 + rowadd + r;
            if (row < N_NODES)
                h[(size_t)row * OUT_DIM + nt * 16 + colbase] = acc[nt][r];
        }
    }
}

// ---------------------------------------------------------------------------
// Kernel 2: per-(node,head) alpha_src/alpha_dst, init max/sum, zero d_out
// ---------------------------------------------------------------------------
__global__ __launch_bounds__(256) void node_alpha_kernel(
        const float* __restrict__ h,
        const float* __restrict__ a_src, const float* __restrict__ a_dst,
        float* __restrict__ alpha_src, float* __restrict__ alpha_dst,
        float* __restrict__ node_max, float* __restrict__ node_sum,
        float* __restrict__ out_acc) {
    const int t = blockIdx.x * blockDim.x + threadIdx.x;
    if (t >= N_NODES * N_HEADS) return;
    const int n = t >> 2, head = t & 3;

    const float4* hv = (const float4*)(h + (size_t)n * OUT_DIM + head * HEAD_DIM);
    const float4* as = (const float4*)(a_src + head * HEAD_DIM);
    const float4* ad = (const float4*)(a_dst + head * HEAD_DIM);
    float4*       ov = (float4*)(out_acc + (size_t)n * OUT_DIM + head * HEAD_DIM);

    float s = 0.f, d = 0.f;
#pragma unroll
    for (int i = 0; i < HEAD_DIM / 4; ++i) {
        float4 hh = hv[i], a1 = as[i], a2 = ad[i];
        s += hh.x * a1.x + hh.y * a1.y + hh.z * a1.z + hh.w * a1.w;
        d += hh.x * a2.x + hh.y * a2.y + hh.z * a2.z + hh.w * a2.w;
        ov[i] = make_float4(0.f, 0.f, 0.f, 0.f);
    }
    alpha_src[t] = s;
    alpha_dst[t] = d;
    node_max[t]  = -1e9f;
    node_sum[t]  = 0.f;
}

// ---------------------------------------------------------------------------
// Kernel 3: edge logits + leaky-relu + segment-max (atomic)
// ---------------------------------------------------------------------------
__global__ __launch_bounds__(256) void edge_logits_kernel(
        const int* __restrict__ ei, const int* __restrict__ etype,
        const float* __restrict__ edge_embed, const float* __restrict__ a_edge,
        const float* __restrict__ alpha_src, const float* __restrict__ alpha_dst,
        float* __restrict__ attn, float* __restrict__ node_max) {
    __shared__ __align__(16) float table[3 * N_HEADS];
    if (threadIdx.x < 3 * N_HEADS) {
        const int ty = threadIdx.x >> 2, hd = threadIdx.x & 3;
        float s = 0.f;
#pragma unroll
        for (int k = 0; k < EE_DIM; ++k)
            s += edge_embed[ty * EE_DIM + k] * a_edge[hd * EE_DIM + k];
        table[ty * N_HEADS + hd] = s;
    }
    __syncthreads();

    const int e = blockIdx.x * blockDim.x + threadIdx.x;
    if (e >= N_EDGES) return;
    const int s  = ei[e];
    const int d  = ei[N_EDGES + e];
    const int ty = etype[e];

    float4 as = *(const float4*)(alpha_src + (size_t)s * 4);
    float4 ad = *(const float4*)(alpha_dst + (size_t)d * 4);
    float4 es = *(const float4*)(table + ty * 4);

    float v[4] = { as.x + ad.x + es.x, as.y + ad.y + es.y,
                   as.z + ad.z + es.z, as.w + ad.w + es.w };
#pragma unroll
    for (int hh = 0; hh < 4; ++hh) {
        float a = v[hh];
        a = a > 0.f ? a : 0.2f * a;       // leaky_relu(0.2)
        v[hh] = a;
        atomicMaxF(node_max + (size_t)d * 4 + hh, a);
    }
    *(float4*)(attn + (size_t)e * 4) = make_float4(v[0], v[1], v[2], v[3]);
}

// ---------------------------------------------------------------------------
// Kernel 4: attn_exp = exp(attn - max[dst]);  segment-sum (atomic)
// ---------------------------------------------------------------------------
__global__ __launch_bounds__(256) void edge_exp_kernel(
        const int* __restrict__ ei, float* __restrict__ attn,
        const float* __restrict__ node_max, float* __restrict__ node_sum) {
    const int e = blockIdx.x * blockDim.x + threadIdx.x;
    if (e >= N_EDGES) return;
    const int d = ei[N_EDGES + e];

    float4 a = *(const float4*)(attn + (size_t)e * 4);
    float4 m = *(const float4*)(node_max + (size_t)d * 4);
    float4 ex;
    ex.x = expf(a.x - m.x);
    ex.y = expf(a.y - m.y);
    ex.z = expf(a.z - m.z);
    ex.w = expf(a.w - m.w);
    *(float4*)(attn + (size_t)e * 4) = ex;

    atomicAddF(node_sum + (size_t)d * 4 + 0, ex.x);
    atomicAddF(node_sum + (size_t)d * 4 + 1, ex.y);
    atomicAddF(node_sum + (size_t)d * 4 + 2, ex.z);
    atomicAddF(node_sum + (size_t)d * 4 + 3, ex.w);
}

// ---------------------------------------------------------------------------
// Kernel 5: weighted scatter-aggregate.  One wave32 per edge:
//   lane L covers out dims [4L,4L+4), head = L>>3; float4 gather of h[src].
// ---------------------------------------------------------------------------
__global__ __launch_bounds__(256) void scatter_kernel(
        const int* __restrict__ ei, const float* __restrict__ attn,
        const float* __restrict__ node_sum, const float* __restrict__ h,
        float* __restrict__ out_acc) {
    const int wave = threadIdx.x >> 5;
    const int lane = threadIdx.x & 31;
    const int e = blockIdx.x * 8 + wave;
    if (e >= N_EDGES) return;

    const int s = ei[e];
    const int d = ei[N_EDGES + e];
    const int head = lane >> 3;

    const float w = attn[(size_t)e * 4 + head] /
                    (node_sum[(size_t)d * 4 + head] + 1e-10f);
    float4 hv = *(const float4*)(h + (size_t)s * OUT_DIM + lane * 4);
    float* op = out_acc + (size_t)d * OUT_DIM + lane * 4;
    atomicAddF(op + 0, hv.x * w);
    atomicAddF(op + 1, hv.y * w);
    atomicAddF(op + 2, hv.z * w);
    atomicAddF(op + 3, hv.w * w);
}

// ---------------------------------------------------------------------------
// Kernel 6: residual + LayerNorm.  One wave32 per node, shfl_xor reduction.
// ---------------------------------------------------------------------------
__global__ __launch_bounds__(256) void ln_kernel(
        const float* __restrict__ h, const float* __restrict__ gamma,
        const float* __restrict__ beta, float* __restrict__ out) {
    const int wave = threadIdx.x >> 5;
    const int lane = threadIdx.x & 31;
    const int n = blockIdx.x * 8 + wave;
    if (n >= N_NODES) return;

    float4 o  = *(const float4*)(out + (size_t)n * OUT_DIM + lane * 4);
    float4 hv = *(const float4*)(h   + (size_t)n * OUT_DIM + lane * 4);
    float4 v  = make_float4(o.x + hv.x, o.y + hv.y, o.z + hv.z, o.w + hv.w);

    float sum = v.x + v.y + v.z + v.w;
    float sq  = v.x * v.x + v.y * v.y + v.z * v.z + v.w * v.w;
#pragma unroll
    for (int m = 16; m >= 1; m >>= 1) {
        sum += __shfl_xor(sum, m, 32);
        sq  += __shfl_xor(sq,  m, 32);
    }
    const float mu  = sum * (1.0f / OUT_DIM);
    const float var = sq * (1.0f / OUT_DIM) - mu * mu;
    const float rs  = rsqrtf(var + LN_EPS);

    float4 g = *(const float4*)(gamma + lane * 4);
    float4 b = *(const float4*)(beta  + lane * 4);
    float4 r;
    r.x = (v.x - mu) * rs * g.x + b.x;
    r.y = (v.y - mu) * rs * g.y + b.y;
    r.z = (v.z - mu) * rs * g.z + b.z;
    r.w = (v.w - mu) * rs * g.w + b.w;
    *(float4*)(out + (size_t)n * OUT_DIM + lane * 4) = r;
}

// ---------------------------------------------------------------------------
// Host-side launcher
// ---------------------------------------------------------------------------
extern "C" void kernel_launch(void* const* d_in, const int* in_sizes, int n_in,
                              void* d_out, int out_size, void* d_ws, size_t ws_size,
                              hipStream_t stream) {
    const float* x          = (const float*)d_in[0];
    const int*   edge_index = (const int*)  d_in[1];
    const int*   edge_type  = (const int*)  d_in[2];
    const float* W          = (const float*)d_in[3];
    const float* a_src      = (const float*)d_in[4];
    const float* a_dst      = (const float*)d_in[5];
    const float* a_edge     = (const float*)d_in[6];
    const float* edge_embed = (const float*)d_in[7];
    const float* ln_gamma   = (const float*)d_in[8];
    const float* ln_beta    = (const float*)d_in[9];
    float*       out        = (float*)d_out;

    // workspace layout (floats)
    float* ws        = (float*)d_ws;
    float* h         = ws;                                     // N*128
    float* alpha_src = h         + (size_t)N_NODES * OUT_DIM;  // N*4
    float* alpha_dst = alpha_src + (size_t)N_NODES * N_HEADS;  // N*4
    float* node_max  = alpha_dst + (size_t)N_NODES * N_HEADS;  // N*4
    float* node_sum  = node_max  + (size_t)N_NODES * N_HEADS;  // N*4
    float* attn      = node_sum  + (size_t)N_NODES * N_HEADS;  // E*4

    // 1) h = x @ W^T  (WMMA)
    gemm_h_kernel<<<(N_NODES + 127) / 128, 256, 0, stream>>>(x, W, h);

    // 2) alpha projections + init reductions + zero d_out
    node_alpha_kernel<<<(N_NODES * N_HEADS + 255) / 256, 256, 0, stream>>>(
        h, a_src, a_dst, alpha_src, alpha_dst, node_max, node_sum, out);

    // 3) edge logits + segment max
    edge_logits_kernel<<<(N_EDGES + 255) / 256, 256, 0, stream>>>(
        edge_index, edge_type, edge_embed, a_edge, alpha_src, alpha_dst,
        attn, node_max);

    // 4) exp + segment sum
    edge_exp_kernel<<<(N_EDGES + 255) / 256, 256, 0, stream>>>(
        edge_index, attn, node_max, node_sum);

    // 5) weighted scatter aggregate (one wave per edge)
    scatter_kernel<<<(N_EDGES + 7) / 8, 256, 0, stream>>>(
        edge_index, attn, node_sum, h, out);

    // 6) residual + LayerNorm (one wave per node)
    ln_kernel<<<(N_NODES + 7) / 8, 256, 0, stream>>>(h, ln_gamma, ln_beta, out);
}